// FeatureEmbedding_42889543418184
// MI455X (gfx1250) — compile-verified
//
#include <hip/hip_runtime.h>
#include <hip/hip_bf16.h>

typedef float v2f __attribute__((ext_vector_type(2)));
typedef float v8f __attribute__((ext_vector_type(8)));
typedef int   v4i __attribute__((ext_vector_type(4)));

#define R_TOT   192000
#define NFLD    7
#define EMB     64
#define K1DIM   448      // 7*64
#define N1DIM   256
#define N2DIM   64
#define KCPAD   136      // 7 + 64 + 64 padded to multiple of 4
#define FMW     72       // fm buffer: cols 0..6 fm1, 7..70 fm2, 71 zero pad
#define TBLROWS 537      // 2+1+1+500+7+24+2
#define BN_EPS  1e-5f
#define KCH1    16       // W1 k-chunk for GEMM1
#define KCH2    16       // Y1 k-chunk for GEMM2
#define APAD    20       // A-tile row stride (floats): 16B aligned, bank-conflict-free

__constant__ int c_base[8] = {0, 2, 3, 4, 504, 511, 535, 537};

// ---- CDNA5 async global->LDS copy (ASYNCcnt) with portable fallback ----
#if defined(__has_builtin)
#if __has_builtin(__builtin_amdgcn_global_load_async_to_lds_b128) && \
    __has_builtin(__builtin_amdgcn_s_wait_asynccnt)
#define USE_ASYNC 1
#endif
#endif

typedef __attribute__((address_space(1))) v4i* gptr4;
typedef __attribute__((address_space(3))) v4i* lptr4;

__device__ __forceinline__ void cp16(float* ldst, const float* gsrc) {
#ifdef USE_ASYNC
    v4i* g0 = (v4i*)gsrc;   // drop const, reinterpret (generic AS)
    v4i* l0 = (v4i*)ldst;
    __builtin_amdgcn_global_load_async_to_lds_b128((gptr4)g0, (lptr4)l0, 0, 0);
#else
    *(float4*)ldst = *(const float4*)gsrc;
#endif
}
__device__ __forceinline__ void async_wait() {
#ifdef USE_ASYNC
    __builtin_amdgcn_s_wait_asynccnt(0);
#endif
}

struct P {
    const float* cov;
    const float* fo[7];
    const float* so[7];
    const float *W1, *b1, *g1, *be1;
    const float *W2, *b2, *g2, *be2;
    const float *Wc, *cb;
    float *SO, *FO, *WcP;
    float *sum1, *sq1, *sc1, *sh1;
    float *sum2, *sq2, *sc2, *sh2;
    float *FM, *Y1, *Y2;
    float *out;
};

// ---------------- init: pack tables, pad concat-W, zero BN accumulators ----------------
__global__ void k_init(P p) {
    int t = blockIdx.x * blockDim.x + threadIdx.x;
    if (t < TBLROWS * EMB) {
        int row = t >> 6, e = t & 63;
        int f = 0;
        #pragma unroll
        for (int i = 1; i < 7; ++i) f += (row >= c_base[i]);
        p.SO[t] = p.so[f][(row - c_base[f]) * EMB + e];
        return;
    }
    t -= TBLROWS * EMB;
    if (t < TBLROWS) {
        int f = 0;
        #pragma unroll
        for (int i = 1; i < 7; ++i) f += (t >= c_base[i]);
        p.FO[t] = p.fo[f][t - c_base[f]];
        return;
    }
    t -= TBLROWS;
    if (t < KCPAD * N2DIM) {
        int r = t >> 6;
        p.WcP[t] = (r < 135) ? p.Wc[t] : 0.0f;
        return;
    }
    t -= KCPAD * N2DIM;
    if (t < N1DIM) { p.sum1[t] = 0.0f; p.sq1[t] = 0.0f; return; }
    t -= N1DIM;
    if (t < N2DIM) { p.sum2[t] = 0.0f; p.sq2[t] = 0.0f; }
}

// ---------------- embed: fm_first + fm_second into FM[R,72] ----------------
__global__ void __launch_bounds__(256) k_embed(P p) {
    long g = (long)blockIdx.x * blockDim.x + threadIdx.x; // R*64 threads
    int row = (int)(g >> 6);
    int e   = (int)(g & 63);
    if (row >= R_TOT) return;
    const float* cr = p.cov + (long)row * NFLD;
    float s = 0.0f, q = 0.0f;
    #pragma unroll
    for (int f = 0; f < NFLD; ++f) {
        float c = cr[f];
        int arow; float v;
        if (f == 1 || f == 2) { arow = c_base[f];            v = c;    }
        else                  { arow = c_base[f] + (int)c;   v = 1.0f; }
        float sv = p.SO[(long)arow * EMB + e] * v;
        s += sv; q += sv * sv;
        if (e == f) p.FM[(long)row * FMW + f] = p.FO[arow] * v; // fm_first
    }
    p.FM[(long)row * FMW + 7 + e] = 0.5f * (s * s - q);         // fm_second
    if (e == 0) p.FM[(long)row * FMW + 71] = 0.0f;              // pad
}

// ---------------- GEMM1: Y1 = X0 @ W1 + b1 (A gathered, B async double-buffered) ------
__global__ void __launch_bounds__(256) k_gemm1(P p) {
    __shared__ int   lidx[32 * 8];
    __shared__ float lval[32 * 8];
    __shared__ __align__(16) float lW[2][KCH1 * N1DIM]; // 2 x 16KB
    __shared__ float ssum[N1DIM];
    __shared__ float ssq [N1DIM];

    const int tid   = threadIdx.x;
    const int rbase = blockIdx.x * 32;

    if (tid < 32 * NFLD) {
        int r = tid / NFLD, f = tid % NFLD;
        float c = p.cov[(long)(rbase + r) * NFLD + f];
        int arow; float v;
        if (f == 1 || f == 2) { arow = c_base[f];          v = c;    }
        else                  { arow = c_base[f] + (int)c; v = 1.0f; }
        lidx[r * 8 + f] = arow;
        lval[r * 8 + f] = v;
    }
    if (tid < N1DIM) { ssum[tid] = 0.0f; ssq[tid] = 0.0f; }

    auto issueW1 = [&](int c, float* dst) {
        const float* g = p.W1 + (size_t)c * KCH1 * N1DIM;
        #pragma unroll
        for (int j = 0; j < (KCH1 * N1DIM / 4) / 256; ++j) { // 4 x b128 per thread
            int idx = (tid + j * 256) * 4;
            cp16(dst + idx, g + idx);
        }
    };

    const int wv = tid >> 5, lane = tid & 31;
    const int wm = wv >> 2, wn = wv & 3;          // 2 M-waves x 4 N-waves
    const int l15 = lane & 15, kh = lane >> 4;
    const int mloc = wm * 16 + l15;
    const int nb   = wn * 64;
    const int aoff = mloc * 8;

    v8f acc[4];
    #pragma unroll
    for (int t = 0; t < 4; ++t) acc[t] = (v8f){0,0,0,0,0,0,0,0};

    const int NCH = K1DIM / KCH1; // 28
    issueW1(0, lW[0]);
    async_wait();
    __syncthreads();

    for (int c = 0; c < NCH; ++c) {
        if (c + 1 < NCH) issueW1(c + 1, lW[(c + 1) & 1]);
        const float* wb = lW[c & 1];
        #pragma unroll
        for (int kc = 0; kc < KCH1; kc += 4) {
            int col = kc + kh * 2;
            int ka  = c * KCH1 + col;
            int f   = ka >> 6, e = ka & 63;
            int   arow = lidx[aoff + f];
            float v    = lval[aoff + f];
            const float* sp = p.SO + (long)arow * EMB + e;
            v2f a; a.x = sp[0] * v; a.y = sp[1] * v;
            #pragma unroll
            for (int t = 0; t < 4; ++t) {
                int n = nb + t * 16 + l15;
                v2f b; b.x = wb[col * N1DIM + n]; b.y = wb[(col + 1) * N1DIM + n];
                acc[t] = __builtin_amdgcn_wmma_f32_16x16x4_f32(
                    false, a, false, b, (short)0, acc[t], false, false);
            }
        }
        async_wait();
        __syncthreads();
    }

    #pragma unroll
    for (int t = 0; t < 4; ++t) {
        int n = nb + t * 16 + l15;
        float bias = p.b1[n];
        float s = 0.0f, q = 0.0f;
        #pragma unroll
        for (int i = 0; i < 8; ++i) {
            float y = acc[t][i] + bias;
            int r = rbase + wm * 16 + kh * 8 + i; // C layout: vgpr i -> row i + 8*kh
            p.Y1[(long)r * N1DIM + n] = y;
            s += y; q += y * y;
        }
        atomicAdd(&ssum[n], s);
        atomicAdd(&ssq [n], q);
    }
    __syncthreads();
    if (tid < N1DIM) {
        atomicAdd(&p.sum1[tid], ssum[tid]);
        atomicAdd(&p.sq1 [tid], ssq [tid]);
    }
}

// ---------------- BN stats -> per-column scale/shift ----------------
__global__ void k_stats(const float* sum, const float* sq, const float* g,
                        const float* be, float* sc, float* sh, int n) {
    int i = blockIdx.x * blockDim.x + threadIdx.x;
    if (i >= n) return;
    float mu  = sum[i] * (1.0f / R_TOT);
    float var = sq[i]  * (1.0f / R_TOT) - mu * mu;
    float inv = rsqrtf(var + BN_EPS);
    float s   = g[i] * inv;
    sc[i] = s;
    sh[i] = be[i] - mu * s;
}

// ---------------- GEMM2: Y2 = relu(bn1(Y1)) @ W2 + b2 (A async, B LDS-resident) -------
__global__ void __launch_bounds__(256) k_gemm2(P p) {
    __shared__ __align__(16) float lW2[N1DIM * N2DIM];   // 64KB, whole W2
    __shared__ __align__(16) float lA[2][128 * APAD];    // 2 x 10KB, padded A tile
    __shared__ float lsc[N1DIM], lsh[N1DIM];
    __shared__ float ssum[N2DIM];
    __shared__ float ssq [N2DIM];

    const int tid = threadIdx.x;
    const int rb0 = blockIdx.x * 128;

    // stage whole W2 + BN params + zero stats
    #pragma unroll
    for (int j = 0; j < (N1DIM * N2DIM / 4) / 256; ++j) { // 16 x b128 per thread
        int idx = (tid + j * 256) * 4;
        cp16(lW2 + idx, p.W2 + idx);
    }
    lsc[tid] = p.sc1[tid];
    lsh[tid] = p.sh1[tid];
    if (tid < N2DIM) { ssum[tid] = 0.0f; ssq[tid] = 0.0f; }

    auto issueA = [&](int c, float* dst) {
        #pragma unroll
        for (int j = 0; j < 2; ++j) {               // 512 b128 segs / 256 threads
            int idx = tid + j * 256;                // 0..511
            int r = idx >> 2, seg = idx & 3;        // row 0..127, 16B segment 0..3
            cp16(dst + r * APAD + seg * 4,
                 p.Y1 + (size_t)(rb0 + r) * N1DIM + c * KCH2 + seg * 4);
        }
    };

    const int wv = tid >> 5, lane = tid & 31;
    const int l15 = lane & 15, kh = lane >> 4;
    const int arow = wv * 16 + l15;                 // local A row

    v8f acc[4];
    #pragma unroll
    for (int t = 0; t < 4; ++t) acc[t] = (v8f){0,0,0,0,0,0,0,0};

    const int NCH = N1DIM / KCH2; // 16
    issueA(0, lA[0]);
    async_wait();
    __syncthreads();

    for (int c = 0; c < NCH; ++c) {
        if (c + 1 < NCH) issueA(c + 1, lA[(c + 1) & 1]);
        const float* ab = lA[c & 1];
        #pragma unroll
        for (int kc = 0; kc < KCH2; kc += 4) {
            int col = kc + kh * 2;
            int ka  = c * KCH2 + col;
            float r0 = ab[arow * APAD + col];
            float r1 = ab[arow * APAD + col + 1];
            v2f a;
            a.x = fmaxf(0.0f, r0 * lsc[ka]     + lsh[ka]);
            a.y = fmaxf(0.0f, r1 * lsc[ka + 1] + lsh[ka + 1]);
            #pragma unroll
            for (int t = 0; t < 4; ++t) {
                int n = t * 16 + l15;
                v2f b; b.x = lW2[ka * N2DIM + n]; b.y = lW2[(ka + 1) * N2DIM + n];
                acc[t] = __builtin_amdgcn_wmma_f32_16x16x4_f32(
                    false, a, false, b, (short)0, acc[t], false, false);
            }
        }
        async_wait();
        __syncthreads();
    }

    #pragma unroll
    for (int t = 0; t < 4; ++t) {
        int n = t * 16 + l15;
        float bias = p.b2[n];
        float s = 0.0f, q = 0.0f;
        #pragma unroll
        for (int i = 0; i < 8; ++i) {
            float y = acc[t][i] + bias;
            int r = rb0 + wv * 16 + kh * 8 + i;
            p.Y2[(long)r * N2DIM + n] = y;          // raw pre-BN
            s += y; q += y * y;
        }
        atomicAdd(&ssum[n], s);
        atomicAdd(&ssq [n], q);
    }
    __syncthreads();
    if (tid < N2DIM) {
        atomicAdd(&p.sum2[tid], ssum[tid]);
        atomicAdd(&p.sq2 [tid], ssq [tid]);
    }
}

// ---------------- GEMM3: out = [fm1|fm2|relu(bn2(Y2))|0] @ Wc + cb -------------------
__device__ __forceinline__ float loadA3(const P& p, long row, int k) {
    if (k < 71) return p.FM[row * FMW + k];
    if (k < 135) {
        int e = k - 71;
        float y = p.Y2[row * N2DIM + e];
        return fmaxf(0.0f, y * p.sc2[e] + p.sh2[e]);
    }
    return 0.0f;
}

__global__ void __launch_bounds__(256) k_gemm3(P p) {
    __shared__ __align__(16) float lWc[KCPAD * N2DIM]; // 34KB, whole padded concat-W
    const int tid = threadIdx.x;

    #pragma unroll
    for (int j = 0; j < 9; ++j) {                    // 2176 b128 segs
        int idx = tid + j * 256;
        if (idx < (KCPAD * N2DIM) / 4) cp16(lWc + idx * 4, p.WcP + idx * 4);
    }
    async_wait();
    __syncthreads();

    const int wv = tid >> 5, lane = tid & 31;
    const int l15 = lane & 15, kh = lane >> 4;
    const int rb = blockIdx.x * 128 + wv * 16;

    v8f acc[4];
    #pragma unroll
    for (int t = 0; t < 4; ++t) acc[t] = (v8f){0,0,0,0,0,0,0,0};

    const long arow = (long)(rb + l15);
    for (int k0 = 0; k0 < KCPAD; k0 += 4) {
        int ka = k0 + kh * 2;
        v2f a; a.x = loadA3(p, arow, ka); a.y = loadA3(p, arow, ka + 1);
        #pragma unroll
        for (int t = 0; t < 4; ++t) {
            int n = t * 16 + l15;
            v2f b; b.x = lWc[ka * N2DIM + n]; b.y = lWc[(ka + 1) * N2DIM + n];
            acc[t] = __builtin_amdgcn_wmma_f32_16x16x4_f32(
                false, a, false, b, (short)0, acc[t], false, false);
        }
    }
    #pragma unroll
    for (int t = 0; t < 4; ++t) {
        int n = t * 16 + l15;
        float bias = p.cb[n];
        #pragma unroll
        for (int i = 0; i < 8; ++i) {
            int r = rb + kh * 8 + i;
            p.out[(long)r * N2DIM + n] = acc[t][i] + bias;
        }
    }
}

extern "C" void kernel_launch(void* const* d_in, const int* in_sizes, int n_in,
                              void* d_out, int out_size, void* d_ws, size_t ws_size,
                              hipStream_t stream) {
    P p{};
    p.cov = (const float*)d_in[0];
    for (int i = 0; i < 7; ++i) p.fo[i] = (const float*)d_in[1 + i];
    for (int i = 0; i < 7; ++i) p.so[i] = (const float*)d_in[8 + i];
    p.W1 = (const float*)d_in[15]; p.b1 = (const float*)d_in[16];
    p.g1 = (const float*)d_in[17]; p.be1 = (const float*)d_in[18];
    p.W2 = (const float*)d_in[19]; p.b2 = (const float*)d_in[20];
    p.g2 = (const float*)d_in[21]; p.be2 = (const float*)d_in[22];
    p.Wc = (const float*)d_in[23]; p.cb = (const float*)d_in[24];

    float* ws = (float*)d_ws;
    size_t o = 0;
    auto alloc = [&](size_t n) { float* r = ws + o; o += (n + 63) & ~(size_t)63; return r; };
    p.SO   = alloc((size_t)TBLROWS * EMB);
    p.FO   = alloc(TBLROWS);
    p.WcP  = alloc((size_t)KCPAD * N2DIM);
    p.sum1 = alloc(N1DIM); p.sq1 = alloc(N1DIM); p.sc1 = alloc(N1DIM); p.sh1 = alloc(N1DIM);
    p.sum2 = alloc(N2DIM); p.sq2 = alloc(N2DIM); p.sc2 = alloc(N2DIM); p.sh2 = alloc(N2DIM);
    p.FM   = alloc((size_t)R_TOT * FMW);
    p.Y1   = alloc((size_t)R_TOT * N1DIM);
    p.Y2   = alloc((size_t)R_TOT * N2DIM);
    p.out  = (float*)d_out;

    int init_tot = TBLROWS * EMB + TBLROWS + KCPAD * N2DIM + N1DIM + N2DIM;
    k_init <<<(init_tot + 255) / 256, 256, 0, stream>>>(p);
    k_embed<<<(R_TOT * EMB) / 256,    256, 0, stream>>>(p);
    k_gemm1<<<R_TOT / 32,             256, 0, stream>>>(p);
    k_stats<<<1, 256, 0, stream>>>(p.sum1, p.sq1, p.g1, p.be1, p.sc1, p.sh1, N1DIM);
    k_gemm2<<<R_TOT / 128,            256, 0, stream>>>(p);
    k_stats<<<1, 64, 0, stream>>>(p.sum2, p.sq2, p.g2, p.be2, p.sc2, p.sh2, N2DIM);
    k_gemm3<<<R_TOT / 128,            256, 0, stream>>>(p);
}